// StandardISAModule_23347442221065
// MI455X (gfx1250) — compile-verified
//
#include <hip/hip_runtime.h>

#define DML 512    // d_model
#define NTOK 2048  // tokens
#define FFD 2048   // d_ff
#define NH 4
#define NL 8

typedef __attribute__((ext_vector_type(2))) float v2f;
typedef __attribute__((ext_vector_type(8))) float v8f;

// C = op(A) @ op(B) with epilogues.
// TA: A accessed as A[k*lda + m] (transposed), else A[m*lda + k]
// TB: B accessed as B[n*ldb + k] (transposed), else B[k*ldb + n]
// EPI: 0 = Cout = acc
//      1 = Cout += acc                      (head accumulation into attn)
//      2 = Cout = relu(acc + bias[row])     (ff1)
//      3 = Cout = acc + bias[row] + Cin     (layer output w/ residual)
//
// Block = 256 threads = 8 waves; block tile 128(M) x 256(N).
// Each wave computes a 64x64 tile = 4x4 WMMA 16x16 tiles (128 acc VGPRs).
template <bool TA, bool TB, int EPI>
__global__ __launch_bounds__(256) void gemm_wmma_f32(
    const float* __restrict__ A, const float* __restrict__ B,
    const float* __restrict__ Cin, const float* __restrict__ bias,
    float* __restrict__ Cout, int K, int lda, int ldb, int ldc)
{
    const int lane = threadIdx.x & 31;
    const int wave = threadIdx.x >> 5;
    const int half = lane >> 4;   // 0: lanes 0-15 (K=kf,kf+1), 1: lanes 16-31 (K=kf+2,kf+3)
    const int lq   = lane & 15;

    const int wr = wave & 1;      // 2 row groups of 64 rows
    const int wc = wave >> 1;     // 4 col groups of 64 cols
    const int mBase = blockIdx.y * 128 + wr * 64;
    const int nBase = blockIdx.x * 256 + wc * 64;

    v8f acc[4][4];
#pragma unroll
    for (int i = 0; i < 4; ++i)
#pragma unroll
        for (int j = 0; j < 4; ++j) acc[i][j] = v8f{};

    for (int kk = 0; kk < K; kk += 4) {
        const int kf = kk + 2 * half;   // even -> 8B-aligned vector loads below
        v2f a[4], b[4];
#pragma unroll
        for (int i = 0; i < 4; ++i) {
            const int row = mBase + i * 16 + lq;
            if (TA) {
                a[i][0] = A[(size_t)kf * lda + row];        // lane-consecutive, coalesced
                a[i][1] = A[(size_t)(kf + 1) * lda + row];
            } else {
                a[i] = *(const v2f*)&A[(size_t)row * lda + kf];  // b64 per lane
            }
        }
#pragma unroll
        for (int j = 0; j < 4; ++j) {
            const int col = nBase + j * 16 + lq;
            if (TB) {
                b[j] = *(const v2f*)&B[(size_t)col * ldb + kf];  // b64 per lane
            } else {
                b[j][0] = B[(size_t)kf * ldb + col];        // lane-consecutive, coalesced
                b[j][1] = B[(size_t)(kf + 1) * ldb + col];
            }
        }
#pragma unroll
        for (int i = 0; i < 4; ++i)
#pragma unroll
            for (int j = 0; j < 4; ++j)
                acc[i][j] = __builtin_amdgcn_wmma_f32_16x16x4_f32(
                    false, a[i], false, b[j], (short)0, acc[i][j], false, false);
    }

#pragma unroll
    for (int i = 0; i < 4; ++i) {
#pragma unroll
        for (int j = 0; j < 4; ++j) {
            const int col = nBase + j * 16 + lq;
#pragma unroll
            for (int r = 0; r < 8; ++r) {
                const int row = mBase + i * 16 + r + 8 * half;
                const size_t idx = (size_t)row * ldc + col;
                float v = acc[i][j][r];
                if (EPI == 0) {
                    Cout[idx] = v;
                } else if (EPI == 1) {
                    Cout[idx] += v;
                } else if (EPI == 2) {
                    v += bias[row];
                    Cout[idx] = v > 0.f ? v : 0.f;
                } else {
                    Cout[idx] = v + bias[row] + Cin[idx];
                }
            }
        }
    }
}

// Row-wise softmax (rows contiguous; Sc = scores^T so this is the reference's
// softmax over axis n).
__global__ __launch_bounds__(256) void softmax_rows(float* __restrict__ S, int ncols)
{
    float* r = S + (size_t)blockIdx.x * ncols;
    __shared__ float red[256];
    const int tid = threadIdx.x;

    float m = -3.4e38f;
    for (int c = tid; c < ncols; c += 256) m = fmaxf(m, r[c]);
    red[tid] = m;
    __syncthreads();
    for (int s = 128; s > 0; s >>= 1) {
        if (tid < s) red[tid] = fmaxf(red[tid], red[tid + s]);
        __syncthreads();
    }
    m = red[0];
    __syncthreads();

    float sum = 0.f;
    for (int c = tid; c < ncols; c += 256) {
        float e = expf(r[c] - m);
        r[c] = e;
        sum += e;
    }
    red[tid] = sum;
    __syncthreads();
    for (int s = 128; s > 0; s >>= 1) {
        if (tid < s) red[tid] += red[tid + s];
        __syncthreads();
    }
    const float inv = 1.0f / red[0];
    for (int c = tid; c < ncols; c += 256) r[c] *= inv;
}

extern "C" void kernel_launch(void* const* d_in, const int* in_sizes, int n_in,
                              void* d_out, int out_size, void* d_ws, size_t ws_size,
                              hipStream_t stream)
{
    const float* X   = (const float*)d_in[0];
    const float* KtQ = (const float*)d_in[1];
    const float* V   = (const float*)d_in[2];
    const float* W1  = (const float*)d_in[3];
    const float* b1  = (const float*)d_in[4];
    const float* W2  = (const float*)d_in[5];
    const float* b2  = (const float*)d_in[6];
    float* out = (float*)d_out;

    float* ws = (float*)d_ws;
    const size_t DN = (size_t)DML * NTOK;       // 1M floats
    float* Xc   = ws;                           //  4 MB  current activations
    float* attn = ws + DN;                      //  4 MB
    float* T    = ws + 2 * DN;                  //  4 MB  KtQ_h @ X
    float* U    = ws + 3 * DN;                  //  4 MB  V_h @ X
    float* Sc   = ws + 4 * DN;                  // 16 MB  scores^T / weights^T
    float* ff1  = Sc + (size_t)NTOK * NTOK;     // 16 MB

    hipMemcpyAsync(Xc, X, DN * sizeof(float), hipMemcpyDeviceToDevice, stream);

    dim3 blk(256);
    dim3 gDN(NTOK / 256, DML / 128);   // 512 x 2048 outputs
    dim3 gNN(NTOK / 256, NTOK / 128);  // 2048 x 2048 outputs
    dim3 gFN(NTOK / 256, FFD / 128);   // 2048 x 2048 outputs (FF rows)

    for (int l = 0; l < NL; ++l) {
        // attn starts as the residual X
        hipMemcpyAsync(attn, Xc, DN * sizeof(float), hipMemcpyDeviceToDevice, stream);

        for (int h = 0; h < NH; ++h) {
            const float* ktq = KtQ + (size_t)(l * NH + h) * DML * DML;
            const float* vh  = V   + (size_t)(l * NH + h) * DML * DML;

            // T = KtQ_h @ X                       [D,N]
            gemm_wmma_f32<false, false, 0><<<gDN, blk, 0, stream>>>(
                ktq, Xc, nullptr, nullptr, T, DML, DML, NTOK, NTOK);
            // Sc[m,n] = sum_d T[d,m] X[d,n]  ( = scores[n,m] ), LAM = 1
            gemm_wmma_f32<true, false, 0><<<gNN, blk, 0, stream>>>(
                T, Xc, nullptr, nullptr, Sc, DML, NTOK, NTOK, NTOK);
            // softmax over n (contiguous rows of Sc)
            softmax_rows<<<NTOK, blk, 0, stream>>>(Sc, NTOK);
            // U = V_h @ X                         [D,N]
            gemm_wmma_f32<false, false, 0><<<gDN, blk, 0, stream>>>(
                vh, Xc, nullptr, nullptr, U, DML, DML, NTOK, NTOK);
            // attn[d,m] += sum_n U[d,n] * Sc[m,n]
            gemm_wmma_f32<false, true, 1><<<gDN, blk, 0, stream>>>(
                U, Sc, nullptr, nullptr, attn, NTOK, NTOK, NTOK, NTOK);
        }

        const float* w1  = W1 + (size_t)l * FFD * DML;
        const float* bb1 = b1 + (size_t)l * FFD;
        const float* w2  = W2 + (size_t)l * DML * FFD;
        const float* bb2 = b2 + (size_t)l * DML;

        // ff1 = relu(W1 @ attn + b1)              [FF,N]
        gemm_wmma_f32<false, false, 2><<<gFN, blk, 0, stream>>>(
            w1, attn, nullptr, bb1, ff1, DML, DML, NTOK, NTOK);
        // Xnext = attn + W2 @ ff1 + b2            [D,N]
        float* Xn = (l == NL - 1) ? out : Xc;
        gemm_wmma_f32<false, false, 3><<<gDN, blk, 0, stream>>>(
            w2, ff1, attn, bb2, Xn, FFD, FFD, NTOK, NTOK);
    }
}